// NGPHashEncoding_10806137716876
// MI455X (gfx1250) — compile-verified
//
#include <hip/hip_runtime.h>
#include <cstdint>

// NGP hash-grid encoding for MI455X (gfx1250, wave32).
//
// Layout: one lane per (point, level) pair.
//   lane t: level = t & 15, point = t >> 4.
// A wave32 = 2 points x 16 levels -> each 8B store instruction writes
// 32 lanes x 8B = 256B *contiguous* output (out is row-major (N, 16*2)).
//
// Cache policy (the MI455X-specific lever):
//   - table (67 MB) fits in the 192 MB L2 -> regular temporal gathers
//     (L2-resident after first touch; graph replays stay warm).
//   - x reads (48 MB) and out writes (512 MB) are streaming, single-touch
//     -> non-temporal (th:TH_LOAD_NT / th:TH_STORE_NT), so the streams
//     don't evict the L2-resident table.
//
// All indexing is 32-bit: 64M threads, byte offsets <= 512 MB, so every
// address is SGPR base + 32-bit VGPR offset (GVS mode) with no 64-bit
// VALU address math.

#define NGP_GRID 512.0f
#define NGP_LOG2_HASH 19
#define NGP_HASH_MASK ((1u << NGP_LOG2_HASH) - 1u)
#define NGP_LEVELS 16u

typedef float v2f __attribute__((ext_vector_type(2)));

__global__ __launch_bounds__(256) void ngp_hash_encode_kernel(
    const float* __restrict__ x,      // (N, 3)
    const float* __restrict__ table,  // (16, 2^19, 2)
    float* __restrict__ out,          // (N, 32)
    unsigned int n_points)
{
    unsigned int t     = blockIdx.x * 256u + threadIdx.x;   // < 2^31
    unsigned int level = t & (NGP_LEVELS - 1u);
    unsigned int p     = t >> 4;
    if (p >= n_points) return;

    // 16 lanes share one point: one b96 NT load, broadcast by the coalescer.
    const float* xp = x + p * 3u;
    float x0 = __builtin_nontemporal_load(xp + 0);
    float x1 = __builtin_nontemporal_load(xp + 1);
    float x2 = __builtin_nontemporal_load(xp + 2);

    // floor(x*512) -> int32; x in [0,1) so values are non-negative.
    uint32_t cx = (uint32_t)(int)floorf(x0 * NGP_GRID);
    uint32_t cy = (uint32_t)(int)floorf(x1 * NGP_GRID);
    uint32_t cz = (uint32_t)(int)floorf(x2 * NGP_GRID);

    // Wrapping uint32 math == JAX int32 low bits; power-of-two floored mod == AND.
    uint32_t h = (cx * 1u) ^ (cy * 2654435761u) ^ (cz * 805459861u);
    h &= NGP_HASH_MASK;

    // Gather one float2 (8B) for this (level, h). Regular temporal load:
    // keep the 67 MB table resident in L2. 32-bit element index (< 2^23).
    unsigned int feat_idx = (level << NGP_LOG2_HASH) | h;
    const v2f* __restrict__ tbl2 = (const v2f*)table;
    v2f f = tbl2[feat_idx];

    // Coalesced streaming store: out[p*32 + level*2 .. +1]; wave writes
    // 256B contiguous. 32-bit element index (< 2^26).
    v2f* __restrict__ out2 = (v2f*)out;
    __builtin_nontemporal_store(f, out2 + (p * NGP_LEVELS + level));
}

extern "C" void kernel_launch(void* const* d_in, const int* in_sizes, int n_in,
                              void* d_out, int out_size, void* d_ws, size_t ws_size,
                              hipStream_t stream) {
    const float* x     = (const float*)d_in[0];   // in_sizes[0] = N * 3
    const float* table = (const float*)d_in[1];   // 16 * 2^19 * 2
    float* out         = (float*)d_out;           // N * 32

    unsigned int n_points = (unsigned int)(in_sizes[0] / 3);
    unsigned long long total_threads = (unsigned long long)n_points * NGP_LEVELS;
    unsigned int block = 256;
    unsigned int grid = (unsigned int)((total_threads + block - 1) / block);

    ngp_hash_encode_kernel<<<grid, block, 0, stream>>>(x, table, out, n_points);
}